// SupConLoss_41790031790302
// MI455X (gfx1250) — compile-verified
//
#include <hip/hip_runtime.h>
#include <math.h>

typedef __attribute__((ext_vector_type(2))) float v2f;
typedef __attribute__((ext_vector_type(4))) float v4f;
typedef __attribute__((ext_vector_type(8))) float v8f;
typedef __attribute__((ext_vector_type(4))) int   v4i;

#define D_DIM 128
#define TILE 16
#define LDS_STRIDE 132   // 128 + 4 pad -> conflict-free ds_load_b64 fragment reads
#define INV_TEMP (1.0f / 0.07f)

typedef __attribute__((address_space(1))) v4i v4i_glb;   // global int4
typedef __attribute__((address_space(3))) v4i v4i_lds;   // LDS int4
typedef __attribute__((address_space(3))) void lds_void;

// ---------------------------------------------------------------------------
// CDNA5 async memory->LDS copy (GLOBAL_LOAD_ASYNC_TO_LDS_B128, ASYNCcnt).
// Builtin when available; inline asm fallback otherwise.
// ---------------------------------------------------------------------------
__device__ __forceinline__ void async_copy_b128(const float* gsrc, float* ldst) {
#if __has_builtin(__builtin_amdgcn_global_load_async_to_lds_b128)
  __builtin_amdgcn_global_load_async_to_lds_b128(
      (v4i_glb*)gsrc, (v4i_lds*)ldst, /*offset=*/0, /*cpol=*/0);
#else
  unsigned lofs = (unsigned)(unsigned long long)(lds_void*)ldst;
  asm volatile("global_load_async_to_lds_b128 %0, %1, off"
               :: "v"(lofs), "v"(gsrc)
               : "memory");
#endif
}

__device__ __forceinline__ void wait_async0() {
#if __has_builtin(__builtin_amdgcn_s_wait_asynccnt)
  __builtin_amdgcn_s_wait_asynccnt(0);
#else
  asm volatile("s_wait_asynccnt 0" ::: "memory");
#endif
}

// ---------------------------------------------------------------------------
// Kernel 1: view-major reorder + L2 row normalization.
// nmat[i][d] = features[b][v][d] / max(||row||, 1e-8), i = v*B + b
// ---------------------------------------------------------------------------
__global__ void __launch_bounds__(256)
supcon_normalize_kernel(const float* __restrict__ features,
                        float* __restrict__ nmat, int Bsz, int N) {
  const int wavesPerBlock = blockDim.x >> 5;
  const int row  = blockIdx.x * wavesPerBlock + (threadIdx.x >> 5);
  const int lane = threadIdx.x & 31;
  if (row >= N) return;
  const int v = row / Bsz;
  const int b = row - v * Bsz;
  const float* src = features + ((size_t)b * 2 + v) * D_DIM;
  v4f x = *(const v4f*)(src + lane * 4);
  float ss = x.x * x.x + x.y * x.y + x.z * x.z + x.w * x.w;
  #pragma unroll
  for (int m = 1; m <= 16; m <<= 1) ss += __shfl_xor(ss, m, 32);
  const float inv = 1.0f / fmaxf(sqrtf(ss), 1e-8f);
  v4f y;
  y.x = x.x * inv; y.y = x.y * inv; y.z = x.z * inv; y.w = x.w * inv;
  *(v4f*)(nmat + (size_t)row * D_DIM + lane * 4) = y;
}

// ---------------------------------------------------------------------------
// Kernel 2: online logsumexp over the 8192x8192 similarity matrix.
// Block = 128 threads = 4 waves; wave w owns rows [blk*64 + w*16, +16).
// A panel (16x128, scaled 1/TEMP) resident in 64 VGPRs for the whole kernel.
// Column tiles double-buffered in LDS via async global->LDS copies; all 4
// waves share each tile. 32x V_WMMA_F32_16X16X4_F32 per 16x16 logits tile.
//
// fp32 WMMA fragment layouts (CDNA5 ISA 7.12.2):
//   A 16x4 : lane<16 -> M=lane,   VGPR j = K=j ;  lane>=16 -> M=lane-16, K=2+j
//   B 4x16 : lane<16 -> N=lane,   VGPR j = K=j ;  lane>=16 -> N=lane-16, K=2+j
//   C 16x16: lane<16 -> N=lane,   VGPR r = M=r ;  lane>=16 -> N=lane-16, M=8+r
// ---------------------------------------------------------------------------
__global__ void __launch_bounds__(128)
supcon_tiles_kernel(const float* __restrict__ nmat,
                    float* __restrict__ partials, int Bsz, int N) {
  __shared__ float btile[2][TILE * LDS_STRIDE];

  const int tid  = threadIdx.x;
  const int lane = tid & 31;
  const int wave = tid >> 5;
  const int rowBase = blockIdx.x * 64 + wave * TILE;

  // ---- Resident A fragments (rows rowBase..rowBase+15), pre-scaled 1/TEMP.
  v2f afrag[32];
  {
    const int m    = lane & 15;
    const int koff = (lane >> 4) * 2;           // 0 for lanes 0-15, 2 for 16-31
    const float* arow = nmat + (size_t)(rowBase + m) * D_DIM;
    #pragma unroll
    for (int k = 0; k < 32; ++k) {
      v2f t = *(const v2f*)(arow + 4 * k + koff);
      afrag[k].x = t.x * INV_TEMP;
      afrag[k].y = t.y * INV_TEMP;
    }
  }

  // Async-stage one 16x128 column tile into LDS buffer `buf`.
  auto stage = [&](int ctile, int buf) {
    const int colBase = ctile * TILE;
    #pragma unroll
    for (int i = 0; i < 4; ++i) {
      const int f = tid + i * 128;              // b128 chunk id in [0,512)
      const int c = f >> 5;                     // tile row (column vector id)
      const int q = f & 31;                     // float4 within row
      async_copy_b128(nmat + (size_t)(colBase + c) * D_DIM + q * 4,
                      &btile[buf][c * LDS_STRIDE + q * 4]);
    }
  };

  float sumExp[8];
  float pLog[8];
  #pragma unroll
  for (int r = 0; r < 8; ++r) { sumExp[r] = 0.0f; pLog[r] = 0.0f; }

  const int gColLane = lane & 15;               // N within tile for C layout
  const int mOff     = (lane >> 4) * 8;         // M offset for C layout
  const int bkoff    = (lane >> 4) * 2;

  const int numColTiles = N / TILE;
  stage(0, 0);                                  // prologue prefetch

  for (int ct = 0; ct < numColTiles; ++ct) {
    wait_async0();                              // this wave's tile landed
    __syncthreads();                            // ...and everyone else's

    if (ct + 1 < numColTiles) stage(ct + 1, (ct + 1) & 1);  // overlap next

    const float* bt = &btile[ct & 1][0];

    // Pre-issue all B fragment reads (ds waits collapse to one).
    v2f bfrag[32];
    #pragma unroll
    for (int k = 0; k < 32; ++k)
      bfrag[k] = *(const v2f*)(bt + gColLane * LDS_STRIDE + 4 * k + bkoff);

    // 16x16 logits tile via 32 fp32 WMMAs over K=128.
    v8f acc = {0.0f, 0.0f, 0.0f, 0.0f, 0.0f, 0.0f, 0.0f, 0.0f};
    #pragma unroll
    for (int k = 0; k < 32; ++k) {
      acc = __builtin_amdgcn_wmma_f32_16x16x4_f32(
          /*neg_a=*/false, afrag[k], /*neg_b=*/false, bfrag[k],
          /*c_mod=*/(short)0, acc, /*reuse_a=*/false, /*reuse_b=*/false);
    }

    // Online epilogue: sum exp (skip diagonal), capture partner logit.
    const int gCol = ct * TILE + gColLane;
    #pragma unroll
    for (int r = 0; r < 8; ++r) {
      const float l = acc[r];                   // already logits (A pre-scaled)
      const int gRow = rowBase + mOff + r;
      const int partner = gRow + ((gRow < Bsz) ? Bsz : -Bsz);
      sumExp[r] += (gCol == gRow) ? 0.0f : __expf(l);
      pLog[r]   += (gCol == partner) ? l : 0.0f;
    }
  }

  // ---- Reduce across the 16 lanes (N dimension) of each half-wave.
  #pragma unroll
  for (int r = 0; r < 8; ++r) {
    #pragma unroll
    for (int m = 1; m <= 8; m <<= 1) {
      sumExp[r] += __shfl_xor(sumExp[r], m, 32);
      pLog[r]   += __shfl_xor(pLog[r], m, 32);
    }
  }
  // Lanes 0 (rows rowBase..+7) and 16 (rows rowBase+8..+15) hold valid sums.
  float part = 0.0f;
  if ((lane & 15) == 0) {
    #pragma unroll
    for (int r = 0; r < 8; ++r)
      part += pLog[r] - __logf(sumExp[r]);
  }
  part += __shfl_xor(part, 16, 32);             // combine the two halves
  if (lane == 0) partials[blockIdx.x * 4 + wave] = part;
}

// ---------------------------------------------------------------------------
// Kernel 3: reduce per-wave partials; loss = -(TEMP/BASE_TEMP) * mean = -sum/N.
// ---------------------------------------------------------------------------
__global__ void __launch_bounds__(512)
supcon_finalize_kernel(const float* __restrict__ partials,
                       float* __restrict__ out, int nPart, int N) {
  __shared__ float s[512];
  const int t = threadIdx.x;
  s[t] = (t < nPart) ? partials[t] : 0.0f;
  __syncthreads();
  #pragma unroll
  for (int stride = 256; stride > 0; stride >>= 1) {
    if (t < stride) s[t] += s[t + stride];
    __syncthreads();
  }
  if (t == 0) *out = -s[0] / (float)N;
}

// ---------------------------------------------------------------------------
extern "C" void kernel_launch(void* const* d_in, const int* in_sizes, int n_in,
                              void* d_out, int out_size, void* d_ws, size_t ws_size,
                              hipStream_t stream) {
  (void)n_in; (void)out_size; (void)ws_size;
  const float* features = (const float*)d_in[0];
  const int total = in_sizes[0];            // B * V * D
  const int Bsz   = total / (2 * D_DIM);    // 4096
  const int N     = 2 * Bsz;                // 8192

  float* nmat     = (float*)d_ws;                       // N*128 floats = 4 MB
  float* partials = nmat + (size_t)N * D_DIM;           // 512 floats

  // 1) normalize: 8 waves / block, 1 wave per row
  supcon_normalize_kernel<<<(N + 7) / 8, 256, 0, stream>>>(features, nmat, Bsz, N);

  // 2) tiled WMMA logsumexp: 64 rows per block (4 waves x 16 rows)
  const int rowBlocks = N / 64;                         // 128
  supcon_tiles_kernel<<<rowBlocks, 128, 0, stream>>>(nmat, partials, Bsz, N);

  // 3) final scalar
  supcon_finalize_kernel<<<1, 512, 0, stream>>>(partials, (float*)d_out,
                                                rowBlocks * 4, N);
}